// GCNEncoder_37263136260510
// MI455X (gfx1250) — compile-verified
//
#include <hip/hip_runtime.h>

typedef __attribute__((ext_vector_type(2))) float v2f;
typedef __attribute__((ext_vector_type(8))) float v8f;

#define NN 100000      // nodes
#define NE 1600000     // edges (excluding self loops)
#define DIM 128
#define LN_EPS 1e-5f

// ---------------------------------------------------------------------------
// Degree / normalization helpers
// ---------------------------------------------------------------------------
__global__ __launch_bounds__(256) void k_fill1(float* __restrict__ deg) {
    int i = blockIdx.x * blockDim.x + threadIdx.x;
    if (i < NN) deg[i] = 1.0f;   // self-loop contributes 1 to every degree
}

__global__ __launch_bounds__(256) void k_degree(float* __restrict__ deg,
                                                const int* __restrict__ dst) {
    int e = blockIdx.x * blockDim.x + threadIdx.x;
    if (e < NE) atomicAdd(&deg[dst[e]], 1.0f);
}

__global__ __launch_bounds__(256) void k_rsqrt_ip(float* __restrict__ d) {
    int i = blockIdx.x * blockDim.x + threadIdx.x;
    if (i < NN) d[i] = rsqrtf(d[i]);   // deg >= 1 always (self loops)
}

// ---------------------------------------------------------------------------
// H = X @ W  via native fp32 WMMA (V_WMMA_F32_16X16X4_F32), wave32.
// One wave computes a 16x16 output tile; 8 waves/block cover a 16x128 strip.
// A-layout: lane(0-15)=row M, vgpr{0,1}=K{0,1}; lanes 16-31 carry K{2,3}.
// B-layout (column-major mirror): lane(0-15)=col N, vgpr{0,1}=K{0,1}; hi K{2,3}.
// C/D: lanes 0-15 -> M=v, lanes 16-31 -> M=v+8, N=lane&15.
// ---------------------------------------------------------------------------
__global__ __launch_bounds__(256) void k_gemm(const float* __restrict__ X,
                                              const float* __restrict__ W,
                                              float* __restrict__ H) {
    const int wave = threadIdx.x >> 5;         // 0..7 -> N tile
    const int lane = threadIdx.x & 31;
    const int m0   = blockIdx.x * 16;
    const int n0   = wave * 16;
    const int r    = lane & 15;
    const int hl   = lane >> 4;                // 0 or 1

    const float* xrow = X + (size_t)(m0 + r) * DIM + 2 * hl;   // + k
    const float* wcol = W + (size_t)(2 * hl) * DIM + (n0 + r); // + k*DIM

    v8f c = {};
#pragma unroll
    for (int k = 0; k < DIM; k += 4) {
        v2f a, b;
        a.x = xrow[k];
        a.y = xrow[k + 1];
        b.x = wcol[k * DIM];
        b.y = wcol[k * DIM + DIM];
        c = __builtin_amdgcn_wmma_f32_16x16x4_f32(false, a, false, b,
                                                  (short)0, c, false, false);
    }

    float* hrow = H + (size_t)(m0 + 8 * hl) * DIM + n0 + r;
#pragma unroll
    for (int v = 0; v < 8; ++v) hrow[v * DIM] = c[v];
}

// ---------------------------------------------------------------------------
// acc[n,:] = dinv[n]^2 * H[n,:] + bias   (self-loop term + bias, overwrites acc)
// ---------------------------------------------------------------------------
__global__ __launch_bounds__(256) void k_selfinit(float* __restrict__ acc,
                                                  const float* __restrict__ H,
                                                  const float* __restrict__ dinv,
                                                  const float* __restrict__ bias) {
    int idx = (blockIdx.x * blockDim.x + threadIdx.x) * 4;
    if (idx >= NN * DIM) return;
    int node = idx >> 7;
    int d    = idx & (DIM - 1);
    float di = dinv[node];
    float nn = di * di;
    float4 h = *(const float4*)(H + idx);
    float4 b = *(const float4*)(bias + d);
    float4 o;
    o.x = nn * h.x + b.x;
    o.y = nn * h.y + b.y;
    o.z = nn * h.z + b.z;
    o.w = nn * h.w + b.w;
    *(float4*)(acc + idx) = o;
}

// ---------------------------------------------------------------------------
// Edge scatter: acc[dst,:] += dinv[src]*dinv[dst] * H[src,:]
// One wave32 per edge. Edge id is derived via readfirstlane so src/dst/dinv
// become wave-uniform scalar (SMEM) loads and nrm lives in an SGPR; the
// vector pipe only carries the float4 gather + 4x global_atomic_add_f32.
// H and acc are L2-resident (51.2 MB each < 192 MB L2).
// ---------------------------------------------------------------------------
__global__ __launch_bounds__(256) void k_scatter(float* __restrict__ acc,
                                                 const float* __restrict__ H,
                                                 const float* __restrict__ dinv,
                                                 const int* __restrict__ src,
                                                 const int* __restrict__ dst) {
    int wv = __builtin_amdgcn_readfirstlane((int)(threadIdx.x >> 5)); // 0..7, SGPR
    int e  = blockIdx.x * 8 + wv;                                     // uniform
    int lane = threadIdx.x & 31;
    if (e >= NE) return;                                              // uniform
    int s = src[e];                 // scalar loads (uniform address)
    int t = dst[e];
    float nrm = dinv[s] * dinv[t];  // uniform
    float4 v = *(const float4*)(H + (size_t)s * DIM + lane * 4);
    float* p = acc + (size_t)t * DIM + lane * 4;
    atomicAdd(p + 0, nrm * v.x);
    atomicAdd(p + 1, nrm * v.y);
    atomicAdd(p + 2, nrm * v.z);
    atomicAdd(p + 3, nrm * v.w);
}

// ---------------------------------------------------------------------------
// LayerNorm (+optional ReLU): one wave per node row, shfl_xor reduction.
// ---------------------------------------------------------------------------
__global__ __launch_bounds__(256) void k_layernorm(const float* __restrict__ acc,
                                                   float* __restrict__ out,
                                                   const float* __restrict__ g,
                                                   const float* __restrict__ bt,
                                                   int relu) {
    int gid  = blockIdx.x * blockDim.x + threadIdx.x;
    int node = gid >> 5;
    int lane = gid & 31;
    if (node >= NN) return;

    float4 v = *(const float4*)(acc + (size_t)node * DIM + lane * 4);
    float s  = v.x + v.y + v.z + v.w;
    float ss = v.x * v.x + v.y * v.y + v.z * v.z + v.w * v.w;
#pragma unroll
    for (int o = 16; o >= 1; o >>= 1) {
        s  += __shfl_xor(s,  o, 32);
        ss += __shfl_xor(ss, o, 32);
    }
    float mu   = s * (1.0f / DIM);
    float var  = ss * (1.0f / DIM) - mu * mu;
    float rinv = rsqrtf(var + LN_EPS);

    float4 gv = *(const float4*)(g  + lane * 4);
    float4 bv = *(const float4*)(bt + lane * 4);
    float4 o4;
    o4.x = (v.x - mu) * rinv * gv.x + bv.x;
    o4.y = (v.y - mu) * rinv * gv.y + bv.y;
    o4.z = (v.z - mu) * rinv * gv.z + bv.z;
    o4.w = (v.w - mu) * rinv * gv.w + bv.w;
    if (relu) {
        o4.x = fmaxf(o4.x, 0.0f);
        o4.y = fmaxf(o4.y, 0.0f);
        o4.z = fmaxf(o4.z, 0.0f);
        o4.w = fmaxf(o4.w, 0.0f);
    }
    *(float4*)(out + (size_t)node * DIM + lane * 4) = o4;
}

// ---------------------------------------------------------------------------
extern "C" void kernel_launch(void* const* d_in, const int* in_sizes, int n_in,
                              void* d_out, int out_size, void* d_ws, size_t ws_size,
                              hipStream_t stream) {
    (void)in_sizes; (void)n_in; (void)out_size; (void)ws_size;

    const float* x   = (const float*)d_in[0];
    const int*   ei  = (const int*)d_in[1];      // [2,E] int32 (JAX x64 off)
    const int*   src = ei;
    const int*   dst = ei + NE;

    const float* Wp[3]  = {(const float*)d_in[2],  (const float*)d_in[6],  (const float*)d_in[10]};
    const float* bp[3]  = {(const float*)d_in[3],  (const float*)d_in[7],  (const float*)d_in[11]};
    const float* gp[3]  = {(const float*)d_in[4],  (const float*)d_in[8],  (const float*)d_in[12]};
    const float* btp[3] = {(const float*)d_in[5],  (const float*)d_in[9],  (const float*)d_in[13]};

    float* out  = (float*)d_out;
    float* ws   = (float*)d_ws;
    float* dinv = ws;                                  // NN floats
    float* P    = ws + (1 << 17);                      // 512 KB offset, 51.2 MB
    float* Q    = P + (size_t)NN * DIM;                // 51.2 MB

    // ---- degrees (dst-only, shared by all layers) ----
    k_fill1   <<<(NN + 255) / 256, 256, 0, stream>>>(dinv);
    k_degree  <<<(NE + 255) / 256, 256, 0, stream>>>(dinv, dst);
    k_rsqrt_ip<<<(NN + 255) / 256, 256, 0, stream>>>(dinv);

    const int gemm_grid    = NN / 16;                      // 6250, exact
    const int init_grid    = (NN * DIM / 4 + 255) / 256;   // 12500, exact
    const int scatter_grid = NE / 8;                       // 200000, exact
    const int ln_grid      = (NN * 32 + 255) / 256;        // 12500, exact

    for (int layer = 0; layer < 3; ++layer) {
        const float* Xin = (layer == 0) ? x : P;           // P dead after GEMM
        k_gemm    <<<gemm_grid,    256, 0, stream>>>(Xin, Wp[layer], Q);
        k_selfinit<<<init_grid,    256, 0, stream>>>(P, Q, dinv, bp[layer]);
        k_scatter <<<scatter_grid, 256, 0, stream>>>(P, Q, dinv, src, dst);
        float* ln_out = (layer == 2) ? out : P;            // in-place LN for 0,1
        k_layernorm<<<ln_grid,     256, 0, stream>>>(P, ln_out, gp[layer], btp[layer],
                                                     (layer < 2) ? 1 : 0);
    }
}